// RNN_84885733638382
// MI455X (gfx1250) — compile-verified
//
#include <hip/hip_runtime.h>
#include <math.h>

// Problem dims (fixed by reference)
#define NB 64
#define TT 512
#define DD 256
#define HH 256

typedef __attribute__((ext_vector_type(2))) float v2f;
typedef __attribute__((ext_vector_type(8))) float v8f;

// ---------------------------------------------------------------------------
// Phase 1: xW[m, h] = x[m, :] . Wx[:, h] + b[h],  m = n*T + t  (M = 32768)
// One 16x16 output tile per wave. K-loop of 64 x V_WMMA_F32_16X16X4_F32,
// split into two accumulators to shorten the WMMA dependency chain.
// ---------------------------------------------------------------------------
__global__ __launch_bounds__(256) void rnn_xproj_kernel(
    const float* __restrict__ x, const float* __restrict__ Wx,
    const float* __restrict__ b, float* __restrict__ xW) {
  const int lane = threadIdx.x & 31;
  const int wave = threadIdx.x >> 5;
  const int tile = blockIdx.x * 8 + wave;    // 32768 tiles total
  const int tM = tile >> 4;                  // 2048 M-tiles
  const int tN = tile & 15;                  // 16  N-tiles
  const int m    = lane & 15;
  const int koff = (lane >> 4) << 1;         // lanes 0-15: K=k,k+1 ; 16-31: K=k+2,k+3

  const float* arow = x + (size_t)(tM * 16 + m) * DD;   // A: row m of x-tile
  const int colB = tN * 16 + m;

  v8f c0 = {}, c1 = {};
#pragma unroll
  for (int k = 0; k < DD; k += 8) {
    // A fragments: contiguous fp32 pairs (8B aligned) -> global_load_b64
    v2f a0 = *(const v2f*)(arow + k + koff);
    v2f a1 = *(const v2f*)(arow + k + 4 + koff);
    // B fragments: Wx is row-major (D,H); column colB, rows kk and kk+1
    v2f b0, b1;
    b0.x = Wx[(size_t)(k + koff) * HH + colB];
    b0.y = Wx[(size_t)(k + koff + 1) * HH + colB];
    b1.x = Wx[(size_t)(k + 4 + koff) * HH + colB];
    b1.y = Wx[(size_t)(k + 4 + koff + 1) * HH + colB];
    c0 = __builtin_amdgcn_wmma_f32_16x16x4_f32(false, a0, false, b0, (short)0, c0, false, false);
    c1 = __builtin_amdgcn_wmma_f32_16x16x4_f32(false, a1, false, b1, (short)0, c1, false, false);
  }
  v8f c = c0 + c1;

  // C/D layout: lane<16 -> M = r, lane>=16 -> M = r+8 ; N = lane&15
  const int colC = tN * 16 + m;
  const int rb = (lane >> 4) << 3;
  const float bias = b[colC];
#pragma unroll
  for (int r = 0; r < 8; ++r) {
    xW[(size_t)(tM * 16 + rb + r) * HH + colC] = c[r] + bias;
  }
}

// ---------------------------------------------------------------------------
// Phase 2: persistent recurrence. 4 workgroups x 16 batch rows.
// Wh (256KB, transposed) + h-chunk live in LDS (CDNA5: 320KB/WGP).
// 16 waves, one 16x16 column-tile of h each; 512 sequential steps.
// ---------------------------------------------------------------------------
#define LSTRIDE 260   // row stride in dwords (pad, keeps 8B alignment)

__global__ __launch_bounds__(512) void rnn_recur_kernel(
    const float* __restrict__ xW, const float* __restrict__ h0,
    const float* __restrict__ Wh, float* __restrict__ out) {
  extern __shared__ float smem[];
  float* WhT  = smem;                    // [HH][LSTRIDE]  WhT[c][k] = Wh[k][c]
  float* hbuf = smem + HH * LSTRIDE;     // [16][LSTRIDE]

  const int tid  = threadIdx.x;
  const int lane = tid & 31;
  const int wave = tid >> 5;             // 0..15 -> output column tile
  const int n0   = blockIdx.x * 16;      // batch-row chunk

  // Stage Wh transposed into LDS (coalesced global reads, one-time cost)
  for (int i = tid; i < HH * HH; i += 512) {
    const int k = i >> 8, cc = i & 255;
    WhT[cc * LSTRIDE + k] = Wh[i];
  }
  // Stage h0 chunk
  for (int i = tid; i < 16 * HH; i += 512) {
    const int r = i >> 8, cc = i & 255;
    hbuf[r * LSTRIDE + cc] = h0[(size_t)(n0 + r) * HH + cc];
  }
  __syncthreads();

  const int m    = lane & 15;
  const int koff = (lane >> 4) << 1;
  const int colC = wave * 16 + m;
  const int rb   = (lane >> 4) << 3;
  const float* Arow = hbuf + m * LSTRIDE;                // A: h rows
  const float* Brow = WhT + (wave * 16 + m) * LSTRIDE;   // B: Wh^T rows

  for (int t = 0; t < TT; ++t) {
    v8f c0 = {}, c1 = {};
#pragma unroll
    for (int k = 0; k < HH; k += 8) {
      v2f a0 = *(const v2f*)(Arow + k + koff);       // ds_load_b64
      v2f b0 = *(const v2f*)(Brow + k + koff);
      v2f a1 = *(const v2f*)(Arow + k + 4 + koff);
      v2f b1 = *(const v2f*)(Brow + k + 4 + koff);
      c0 = __builtin_amdgcn_wmma_f32_16x16x4_f32(false, a0, false, b0, (short)0, c0, false, false);
      c1 = __builtin_amdgcn_wmma_f32_16x16x4_f32(false, a1, false, b1, (short)0, c1, false, false);
    }
    v8f c = c0 + c1;

    __syncthreads();   // all hbuf reads for this step complete
#pragma unroll
    for (int r = 0; r < 8; ++r) {
      const int rowC = rb + r;
      const size_t gi = (size_t)(n0 + rowC) * (TT * HH) + (size_t)t * HH + colC;
      const float v = tanhf(c[r] + __builtin_nontemporal_load(xW + gi));
      __builtin_nontemporal_store(v, out + gi);      // streamed, never re-read
      hbuf[rowC * LSTRIDE + colC] = v;               // next step's A operand
    }
    __syncthreads();   // new h visible to all waves
  }
}

// ---------------------------------------------------------------------------
// Inputs (order): x(N,T,D) f32, h0(N,H) f32, Wx(D,H) f32, Wh(H,H) f32, b(H) f32
// Output: (N,T,H) f32. Workspace: xW buffer (N*T*H f32 = 33.5 MB).
// ---------------------------------------------------------------------------
extern "C" void kernel_launch(void* const* d_in, const int* in_sizes, int n_in,
                              void* d_out, int out_size, void* d_ws, size_t ws_size,
                              hipStream_t stream) {
  const float* x  = (const float*)d_in[0];
  const float* h0 = (const float*)d_in[1];
  const float* Wx = (const float*)d_in[2];
  const float* Wh = (const float*)d_in[3];
  const float* b  = (const float*)d_in[4];
  float* out = (float*)d_out;
  float* xW  = (float*)d_ws;   // (N*T) x H staging

  // Phase 1: 32768 tiles, 8 waves/WG -> 4096 workgroups
  rnn_xproj_kernel<<<4096, 256, 0, stream>>>(x, Wx, b, xW);

  // Phase 2: 4 persistent workgroups, ~283 KB dynamic LDS each
  const size_t smem_bytes = (size_t)(HH + 16) * LSTRIDE * sizeof(float);
  hipFuncSetAttribute((const void*)rnn_recur_kernel,
                      hipFuncAttributeMaxDynamicSharedMemorySize, (int)smem_bytes);
  rnn_recur_kernel<<<NB / 16, 512, smem_bytes, stream>>>(xW, h0, Wh, out);
}